// ann2_snn1_54760833024461
// MI455X (gfx1250) — compile-verified
//
#include <hip/hip_runtime.h>
#include <math.h>

typedef __attribute__((ext_vector_type(2))) float v2f;
typedef __attribute__((ext_vector_type(8))) float v8f;

// Computes C[M,N] = act(A[M,K] @ W[N,K]^T + bias) with V_WMMA_F32_16X16X4_F32.
// One wave32 per 16x16 output tile. ACT: 0=none, 1=relu, 2=sigmoid.
template <int ACT>
__global__ __launch_bounds__(256) void gemm_bias_act_wmma(
    const float* __restrict__ A,     // [M,K] row-major
    const float* __restrict__ W,     // [N,K] row-major
    const float* __restrict__ bias,  // [N] or nullptr
    float* __restrict__ C,           // [M,N]
    int M, int N, int K, int tiles_m, int tiles_n)
{
  const int gtid = blockIdx.x * blockDim.x + threadIdx.x;
  const int wave = gtid >> 5;
  const int lane = threadIdx.x & 31;
  if (wave >= tiles_m * tiles_n) return;   // wave-uniform: EXEC stays all-1s
  const int tm = wave % tiles_m;           // adjacent waves share the W tile
  const int tn = wave / tiles_m;
  const int m0 = tm << 4;
  const int n0 = tn << 4;

  // ISA 7.12.2 fragment layout for 32-bit 16x4 A (and 4x16 B, transposed):
  // lanes 0-15 carry K={0,1}, lanes 16-31 carry K={2,3}; row/col = lane&15.
  const int  lh    = lane & 15;
  const int  koff  = (lane < 16) ? 0 : 2;
  const int  arow  = m0 + lh;                 // valid: M % 16 == 0 here
  const int  wrow  = n0 + lh;                 // output column / W row
  const bool nvalid = (wrow < N);
  const int  wrow_c = nvalid ? wrow : (N - 1); // clamp: garbage cols never stored

  const float* __restrict__ Ap = A + (size_t)arow * K + koff;
  const float* __restrict__ Wp = W + (size_t)wrow_c * K + koff;

  v8f acc = {};
#pragma unroll 4
  for (int k = 0; k < K; k += 4) {
    v2f a = *(const v2f*)(Ap + k);   // 8B aligned: K % 4 == 0, koff in {0,2}
    v2f b = *(const v2f*)(Wp + k);
    acc = __builtin_amdgcn_wmma_f32_16x16x4_f32(
        /*neg_a=*/false, a, /*neg_b=*/false, b,
        /*c_mod=*/(short)0, acc, /*reuse_a=*/false, /*reuse_b=*/false);
  }

  const float bv    = (bias != nullptr && nvalid) ? bias[wrow_c] : 0.0f;
  const int   rbase = m0 + ((lane < 16) ? 0 : 8);
#pragma unroll
  for (int v = 0; v < 8; ++v) {
    float x = acc[v] + bv;
    if (ACT == 1) x = fmaxf(x, 0.0f);
    if (ACT == 2) x = 1.0f / (1.0f + expf(-x));
    if (nvalid) C[(size_t)(rbase + v) * N + wrow] = x;
  }
}

// LIF membrane recurrence. psp[t] = k_t * drive (scalar k_t since the IIR is
// linear with constant input), so cur[t] = k_t * D + b3 with D = drive @ w3^T.
__global__ __launch_bounds__(256) void snn_recurrence(
    const float* __restrict__ D,   // [1024,10] (no bias)
    const float* __restrict__ b3,  // [10]
    float* __restrict__ out,       // [1024,10,100]
    float a1, float a2, float sigma)
{
  const int idx = blockIdx.x * blockDim.x + threadIdx.x;
  if (idx >= 1024 * 10) return;
  const float d  = D[idx];
  const float bj = b3[idx % 10];
  float k1 = 0.0f, k2 = 0.0f, v = 0.0f, s = 0.0f;
  float* __restrict__ o = out + (size_t)idx * 100;
#pragma unroll 5
  for (int t = 0; t < 100; ++t) {
    const float kt  = a1 * k1 + a2 * k2 + 1.0f;   // dual-exp IIR impulse sum
    k2 = k1; k1 = kt;
    const float cur = kt * d + bj;
    v = sigma * v * (1.0f - s) + cur;             // reset-to-zero on spike
    s = (v >= 1.0f) ? 1.0f : 0.0f;
    o[t] = s;
  }
}

extern "C" void kernel_launch(void* const* d_in, const int* in_sizes, int n_in,
                              void* d_out, int out_size, void* d_ws, size_t ws_size,
                              hipStream_t stream) {
  const float* X  = (const float*)d_in[0];  // [1024,784]
  const float* w1 = (const float*)d_in[1];  // [500,784]
  const float* b1 = (const float*)d_in[2];  // [500]
  const float* w2 = (const float*)d_in[3];  // [500,500]
  const float* b2 = (const float*)d_in[4];  // [500]
  const float* w3 = (const float*)d_in[5];  // [10,500]
  const float* b3 = (const float*)d_in[6];  // [10]

  float* h     = (float*)d_out;                 // [1024,500] staged in d_out (2MB<=4MB)
  float* drive = (float*)d_ws;                  // [1024,500]
  float* Dm    = (float*)d_ws + 1024 * 500;     // [1024,10]
  float* out   = (float*)d_out;                 // final [1024,10,100], fully rewritten

  const double em = exp(-1.0 / 4.0), es = exp(-1.0);
  const float a1    = (float)(em + es);
  const float a2    = (float)(-em * es);
  const float sigma = (float)em;

  // h = relu(X @ w1^T + b1): 64x32 tiles -> 2048 waves -> 256 blocks
  gemm_bias_act_wmma<1><<<(64 * 32 * 32) / 256, 256, 0, stream>>>(
      X, w1, b1, h, 1024, 500, 784, 64, 32);
  // drive = sigmoid(h @ w2^T + b2)
  gemm_bias_act_wmma<2><<<(64 * 32 * 32) / 256, 256, 0, stream>>>(
      h, w2, b2, drive, 1024, 500, 500, 64, 32);
  // D = drive @ w3^T (bias applied per-timestep in the recurrence)
  gemm_bias_act_wmma<0><<<(64 * 1 * 32) / 256, 256, 0, stream>>>(
      drive, w3, nullptr, Dm, 1024, 10, 500, 64, 1);
  // spikes
  snn_recurrence<<<(1024 * 10 + 255) / 256, 256, 0, stream>>>(
      Dm, b3, out, a1, a2, sigma);
}